// TableBatchedEmbeddingBags_48567490183509
// MI455X (gfx1250) — compile-verified
//
#include <hip/hip_runtime.h>
#include <stdint.h>

#define WAVE_SZ 32
#define CH      16              // rows per chunk (one stage buffer = 8 KB)
#define NBUF    2               // double-buffered stages
#define WPB     4               // waves per block
#define TPB     (WPB * WAVE_SZ) // 128 threads

typedef int v4i __attribute__((vector_size(4 * sizeof(int))));
typedef v4i __attribute__((address_space(1))) *g4p; // global int4*
typedef v4i __attribute__((address_space(3))) *l4p; // LDS int4*

#if __has_builtin(__builtin_amdgcn_global_load_async_to_lds_b128)
#define ASYNC_BUILTIN 1
#else
#define ASYNC_BUILTIN 0
#endif

// Async DMA of 16 bytes/lane from global to LDS (whole wave covers one 512B row).
__device__ __forceinline__ void async_load_16B(const float* gsrc, float4* ldst) {
#if ASYNC_BUILTIN
    __builtin_amdgcn_global_load_async_to_lds_b128(
        (g4p)gsrc, (l4p)ldst, /*imm offset*/ 0, /*cpol*/ 0);
#else
    uint32_t loff = (uint32_t)(uintptr_t)ldst;
    asm volatile("global_load_async_to_lds_b128 %0, %1, off"
                 :: "v"(loff), "v"(gsrc)
                 : "memory");
#endif
}

// Wait until at most N async ops outstanding (N must be a compile-time constant).
template <int N>
__device__ __forceinline__ void wait_async() {
#if __has_builtin(__builtin_amdgcn_s_wait_asynccnt)
    __builtin_amdgcn_s_wait_asynccnt(N);
#else
    asm volatile("s_wait_asynccnt %0" :: "i"(N) : "memory");
#endif
    asm volatile("" ::: "memory"); // compiler fence: LDS reads stay after the wait
}

__global__ __launch_bounds__(TPB)
void tbe_pool_kernel(const float* __restrict__ weights,
                     const int*   __restrict__ table_offsets,
                     const int*   __restrict__ indices,
                     const int*   __restrict__ offsets,
                     float*       __restrict__ out,
                     int T, int B, int D /* == 128 */) {
    __shared__ float4 stage[WPB][NBUF][CH][WAVE_SZ]; // 64 KB / block

    const int lane = threadIdx.x & (WAVE_SZ - 1);
    const int wv   = threadIdx.x >> 5;
    const int seg  = blockIdx.x * WPB + wv;     // bag id = t*B + b (table-major)
    const int nbags = T * B;
    if (seg >= nbags) return;

    const int t   = seg / B;
    const int b   = seg - t * B;
    const int beg = offsets[seg];
    const int end = offsets[seg + 1];
    const long long tbase = (long long)table_offsets[t];

    float4 acc = make_float4(0.f, 0.f, 0.f, 0.f);

    const int nIdx = end - beg;
    const int nch  = (nIdx + CH - 1) / CH;   // number of 16-row chunks

    if (nch > 0) {
        // preload chunk 0's indices
        int curIdx = 0;
        {
            const int cnt0 = min(CH, nIdx);
            if (lane < cnt0) curIdx = indices[beg + lane];
        }

        for (int c = 0; c < nch; ++c) {
            const int base = beg + c * CH;
            const int cnt  = min(CH, end - base);
            const int buf  = c & 1;

            // ---- issue async gathers for chunk c ----
            if (cnt == CH) {
#pragma unroll
                for (int j = 0; j < CH; ++j) {
                    const int row = __shfl(curIdx, j, WAVE_SZ);
                    const float* src = weights
                        + (unsigned long long)(tbase + row) * (unsigned)D
                        + (lane << 2);
                    async_load_16B(src, &stage[wv][buf][j][lane]);
                }
            } else { // only possible for the final chunk
                for (int j = 0; j < cnt; ++j) {
                    const int row = __shfl(curIdx, j, WAVE_SZ);
                    const float* src = weights
                        + (unsigned long long)(tbase + row) * (unsigned)D
                        + (lane << 2);
                    async_load_16B(src, &stage[wv][buf][j][lane]);
                }
            }

            // ---- prefetch chunk c+1's indices (overlaps with wait/consume) ----
            int nxtIdx = 0;
            const int nbase = base + CH;
            if (nbase < end) {
                const int ncnt = min(CH, end - nbase);
                if (lane < ncnt) nxtIdx = indices[nbase + lane];
            }

            // ---- consume chunk c-1 (always a full chunk) ----
            if (c > 0) {
                if (cnt == CH) wait_async<CH>(); // newest <=16 pending are chunk c's
                else           wait_async<0>();  // tail chunk short: drain fully
#pragma unroll
                for (int j = 0; j < CH; ++j) {
                    const float4 v = stage[wv][buf ^ 1][j][lane];
                    acc.x += v.x; acc.y += v.y; acc.z += v.z; acc.w += v.w;
                }
                asm volatile("" ::: "memory");
            }
            curIdx = nxtIdx;
        }

        // ---- drain: consume the last chunk ----
        wait_async<0>();
        {
            const int lastCnt = nIdx - (nch - 1) * CH;
            const int lastBuf = (nch - 1) & 1;
            for (int j = 0; j < lastCnt; ++j) {
                const float4 v = stage[wv][lastBuf][j][lane];
                acc.x += v.x; acc.y += v.y; acc.z += v.z; acc.w += v.w;
            }
        }
    }

    // output layout (B, T, D): bag (t,b) -> row b*T + t
    float4* dst = reinterpret_cast<float4*>(
                      out + (unsigned long long)(b * T + t) * (unsigned)D) + lane;
    *dst = acc;
}

extern "C" void kernel_launch(void* const* d_in, const int* in_sizes, int n_in,
                              void* d_out, int out_size, void* d_ws, size_t ws_size,
                              hipStream_t stream) {
    const float* weights       = (const float*)d_in[0];
    const int*   table_offsets = (const int*)d_in[1];
    const int*   indices       = (const int*)d_in[2];
    const int*   offsets       = (const int*)d_in[3];
    float*       out           = (float*)d_out;

    const int T     = in_sizes[1];          // 8
    const int nbags = in_sizes[3] - 1;      // T*B
    const int B     = nbags / T;            // 4096
    const int D     = out_size / nbags;     // 128

    const int blocks = (nbags + WPB - 1) / WPB;
    tbe_pool_kernel<<<dim3(blocks), dim3(TPB), 0, stream>>>(
        weights, table_offsets, indices, offsets, out, T, B, D);
}